// GraphAutoencoder_69982197121234
// MI455X (gfx1250) — compile-verified
//
#include <hip/hip_runtime.h>

#define N_NODES 12288
#define N_EDGES 393216
#define D_IN    128

typedef float v2f __attribute__((ext_vector_type(2)));
typedef float v8f __attribute__((ext_vector_type(8)));

// ---------------------------------------------------------------------------
// ws layout (floats):
//  [0,N)        deg -> dinv          (init 1.0f for self loop)
//  [N,17N)      a1  (agg of p1, 16)  (init 0)
//  [17N,25N)    a2  (agg of p2, 8)   (init 0)
//  [25N,33N)    agz (agg of z,  8)   (init 0)
//  [33N,49N)    p1 = x@W1
//  [49N,65N)    z1 = relu(a1+b1)
//  [65N,73N)    p2 = z1@W2
//  [73N,81N)    z  = relu(a2+b2)
//  [81N,89N)    zs = relu(agz@Ws+bs)
// ---------------------------------------------------------------------------

__global__ void init_ws_kernel(float* __restrict__ ws) {
    int i = blockIdx.x * blockDim.x + threadIdx.x;
    const int total = 33 * N_NODES;
    if (i < total) ws[i] = (i < N_NODES) ? 1.0f : 0.0f;   // deg starts at 1 (self loop)
}

__global__ void deg_accum_kernel(const int* __restrict__ dst, float* __restrict__ deg) {
    int e = blockIdx.x * blockDim.x + threadIdx.x;
    if (e < N_EDGES) atomicAdd(&deg[dst[e]], 1.0f);
}

__global__ void deg_rsqrt_kernel(float* __restrict__ deg) {
    int i = blockIdx.x * blockDim.x + threadIdx.x;
    if (i < N_NODES) deg[i] = rsqrtf(deg[i]);             // deg >= 1 always
}

// out[n,M] = in[n,K] @ W[K,M] (+bias) (relu optional); one thread per element
template <int K, int M, bool BIAS, bool RELU>
__global__ void gemm_small_kernel(const float* __restrict__ in, const float* __restrict__ W,
                                  const float* __restrict__ bias, float* __restrict__ out) {
    int t = blockIdx.x * blockDim.x + threadIdx.x;
    if (t >= N_NODES * M) return;
    int row = t / M, col = t % M;
    float acc = BIAS ? bias[col] : 0.0f;
    const float* ip = in + row * K;
    const float* wp = W + col;
#pragma unroll 8
    for (int k = 0; k < K; ++k) acc = fmaf(ip[k], wp[k * M], acc);
    out[t] = RELU ? fmaxf(acc, 0.0f) : acc;
}

// out[dst] += dinv[src]*dinv[dst] * in[src]  over E edges + N self loops
template <int DIM>
__global__ void agg_kernel(const float* __restrict__ in, float* __restrict__ out,
                           const int* __restrict__ src, const int* __restrict__ dst,
                           const float* __restrict__ dinv) {
    int t = blockIdx.x * blockDim.x + threadIdx.x;
    const int total = (N_EDGES + N_NODES) * DIM;
    if (t >= total) return;
    int e = t / DIM, c = t % DIM;
    int s, d;
    if (e < N_EDGES) { s = src[e]; d = dst[e]; }
    else             { s = d = e - N_EDGES; }             // self loop
    float w = dinv[s] * dinv[d];
    atomicAdd(&out[d * DIM + c], w * in[s * DIM + c]);
}

template <int M>
__global__ void bias_relu_kernel(const float* __restrict__ a, const float* __restrict__ b,
                                 float* __restrict__ out) {
    int t = blockIdx.x * blockDim.x + threadIdx.x;
    if (t < N_NODES * M) out[t] = fmaxf(a[t] + b[t % M], 0.0f);
}

// ---------------------------------------------------------------------------
// a_hat = zs @ zs^T via V_WMMA_F32_16X16X4_F32, 2 WMMAs per 16x16 tile (K=8).
// One wave per tile. Store-bandwidth bound (604 MB of fp32 out).
// f32 A 16x4 layout: lane m(0-15): v0=K0,v1=K1 ; lane m+16: v0=K2,v1=K3.
// B 4x16 symmetric; for zs*zs^T both operands load with identical pattern.
// C/D: VGPR r -> row = r + 8*(lane>>4), col = lane&15.
// ---------------------------------------------------------------------------
__global__ void __launch_bounds__(128)
ahat_wmma_kernel(const float* __restrict__ zs, float* __restrict__ out) {
    const int lane = threadIdx.x;              // 0..31
    const int ti   = blockIdx.y * 4 + threadIdx.y;   // row tile 0..767
    const int tj   = blockIdx.x;                     // col tile 0..767
    const int i0 = ti * 16, j0 = tj * 16;
    const int m  = lane & 15;
    const int kh = lane >> 4;                  // 0 or 1

    const float* ap = zs + (i0 + m) * 8 + 2 * kh;
    const float* bp = zs + (j0 + m) * 8 + 2 * kh;
    v2f a0 = *(const v2f*)(ap);        // K = {2kh, 2kh+1}
    v2f a1 = *(const v2f*)(ap + 4);    // K = {4+2kh, 5+2kh}
    v2f b0 = *(const v2f*)(bp);
    v2f b1 = *(const v2f*)(bp + 4);

    v8f c = {};
    c = __builtin_amdgcn_wmma_f32_16x16x4_f32(false, a0, false, b0, (short)0, c, false, false);
    c = __builtin_amdgcn_wmma_f32_16x16x4_f32(false, a1, false, b1, (short)0, c, false, false);

    float* o = out + (size_t)(i0 + 8 * kh) * N_NODES + j0 + m;
#pragma unroll
    for (int r = 0; r < 8; ++r)
        o[(size_t)r * N_NODES] = c[r];
}

// ---------------------------------------------------------------------------
extern "C" void kernel_launch(void* const* d_in, const int* in_sizes, int n_in,
                              void* d_out, int out_size, void* d_ws, size_t ws_size,
                              hipStream_t stream) {
    const float* x  = (const float*)d_in[0];
    const int*   ei = (const int*)d_in[1];   // [2,E]: row0 = src, row1 = dst
    const float* W1 = (const float*)d_in[2];
    const float* b1 = (const float*)d_in[3];
    const float* W2 = (const float*)d_in[4];
    const float* b2 = (const float*)d_in[5];
    const float* Wa = (const float*)d_in[6];
    const float* ba = (const float*)d_in[7];
    const float* Ws = (const float*)d_in[8];
    const float* bs = (const float*)d_in[9];
    const int* src = ei;
    const int* dst = ei + N_EDGES;

    float* ws  = (float*)d_ws;
    float* deg = ws;                       // N  (becomes dinv)
    float* a1  = ws +      N_NODES;        // 16N
    float* a2  = ws + 17 * N_NODES;        // 8N
    float* agz = ws + 25 * N_NODES;        // 8N
    float* p1  = ws + 33 * N_NODES;        // 16N
    float* z1  = ws + 49 * N_NODES;        // 16N
    float* p2  = ws + 65 * N_NODES;        // 8N
    float* z   = ws + 73 * N_NODES;        // 8N
    float* zs  = ws + 81 * N_NODES;        // 8N

    float* xhat = (float*)d_out;                       // [N,128]
    float* ahat = (float*)d_out + N_NODES * D_IN;      // [N,N]

    const int B = 256;
    auto nb = [](int n, int b) { return (n + b - 1) / b; };

    // degrees + normalization
    init_ws_kernel<<<nb(33 * N_NODES, B), B, 0, stream>>>(ws);
    deg_accum_kernel<<<nb(N_EDGES, B), B, 0, stream>>>(dst, deg);
    deg_rsqrt_kernel<<<nb(N_NODES, B), B, 0, stream>>>(deg);

    // layer 1: p1 = x@W1 ; a1 = A@p1 ; z1 = relu(a1+b1)
    gemm_small_kernel<128, 16, false, false><<<nb(N_NODES * 16, B), B, 0, stream>>>(x, W1, nullptr, p1);
    agg_kernel<16><<<nb((N_EDGES + N_NODES) * 16, B), B, 0, stream>>>(p1, a1, src, dst, deg);
    bias_relu_kernel<16><<<nb(N_NODES * 16, B), B, 0, stream>>>(a1, b1, z1);

    // layer 2: p2 = z1@W2 ; a2 = A@p2 ; z = relu(a2+b2)
    gemm_small_kernel<16, 8, false, false><<<nb(N_NODES * 8, B), B, 0, stream>>>(z1, W2, nullptr, p2);
    agg_kernel<8><<<nb((N_EDGES + N_NODES) * 8, B), B, 0, stream>>>(p2, a2, src, dst, deg);
    bias_relu_kernel<8><<<nb(N_NODES * 8, B), B, 0, stream>>>(a2, b2, z);

    // shared aggregation for both decoder heads: agz = A@z
    agg_kernel<8><<<nb((N_EDGES + N_NODES) * 8, B), B, 0, stream>>>(z, agz, src, dst, deg);

    // x_hat = agz@Wa + ba ; zs = relu(agz@Ws + bs)
    gemm_small_kernel<8, 128, true, false><<<nb(N_NODES * 128, B), B, 0, stream>>>(agz, Wa, ba, xhat);
    gemm_small_kernel<8, 8, true, true><<<nb(N_NODES * 8, B), B, 0, stream>>>(agz, Ws, bs, zs);

    // a_hat = zs @ zs^T  (WMMA f32 16x16x4, store-bound)
    dim3 blk(32, 4);
    dim3 grd(N_NODES / 16, N_NODES / 16 / 4);   // 768 x 192 blocks, 1 wave per tile
    ahat_wmma_kernel<<<grd, blk, 0, stream>>>(zs, ahat);
}